// OrthogonalityLoss_747324309671
// MI455X (gfx1250) — compile-verified
//
#include <hip/hip_runtime.h>

typedef __attribute__((ext_vector_type(2))) float v2f;
typedef __attribute__((ext_vector_type(8))) float v8f;

static constexpr int C   = 4096;  // channels (fixed by module)
static constexpr int TPB = 256;   // 8 wave32 per block

// ---------------------------------------------------------------------------
// Pass 1: stream input [B][C][2] (fp32, 128 MiB) once with float4 loads.
// Each thread owns 2 channels (one float4 column) and accumulates the 5
// statistics {Sx0, Sx1, Sx0x0, Sx0x1, Sx1x1} in registers over its slice of
// rows. Deterministic: fixed summation order, no atomics.
// partial layout: [NY][5][C] floats.
// ---------------------------------------------------------------------------
__global__ void __launch_bounds__(TPB)
orth_pass1(const float4* __restrict__ in, float* __restrict__ partial,
           int rowsPerBlock) {
  const int W   = C / 2;                                // float4 per row
  const int col = blockIdx.x * TPB + threadIdx.x;       // float4 column
  const int gy  = blockIdx.y;                           // b-slice

  const float4* p = in + (size_t)gy * rowsPerBlock * W + col;

  float s0a = 0.f, s1a = 0.f, m00a = 0.f, m01a = 0.f, m11a = 0.f;
  float s0b = 0.f, s1b = 0.f, m00b = 0.f, m01b = 0.f, m11b = 0.f;

#pragma unroll 4
  for (int r = 0; r < rowsPerBlock; ++r) {
    float4 v = p[(size_t)r * W];  // (x0,x1) of channel 2*col, then 2*col+1
    s0a += v.x; s1a += v.y;
    m00a = fmaf(v.x, v.x, m00a);
    m01a = fmaf(v.x, v.y, m01a);
    m11a = fmaf(v.y, v.y, m11a);
    s0b += v.z; s1b += v.w;
    m00b = fmaf(v.z, v.z, m00b);
    m01b = fmaf(v.z, v.w, m01b);
    m11b = fmaf(v.w, v.w, m11b);
  }

  float* base = partial + (size_t)gy * 5 * C;
  ((float2*)(base + 0 * C))[col] = make_float2(s0a,  s0b);
  ((float2*)(base + 1 * C))[col] = make_float2(s1a,  s1b);
  ((float2*)(base + 2 * C))[col] = make_float2(m00a, m00b);
  ((float2*)(base + 3 * C))[col] = make_float2(m01a, m01b);
  ((float2*)(base + 4 * C))[col] = make_float2(m11a, m11b);
}

// ---------------------------------------------------------------------------
// Pass 2: one thread per channel. Fold the NY b-slice partials (coalesced:
// consecutive threads read consecutive channels), then form the centered
// covariance and the per-channel loss  sqrt(2)*|cov01| / sqrt(cov00*cov11).
// ---------------------------------------------------------------------------
__global__ void __launch_bounds__(TPB)
orth_pass2(const float* __restrict__ partial, float* __restrict__ fro,
           int NY, float invB) {
  const int c = blockIdx.x * TPB + threadIdx.x;
  float S0 = 0.f, S1 = 0.f, M00 = 0.f, M01 = 0.f, M11 = 0.f;
  for (int gy = 0; gy < NY; ++gy) {
    const float* base = partial + (size_t)gy * 5 * C;
    S0  += base[0 * C + c];
    S1  += base[1 * C + c];
    M00 += base[2 * C + c];
    M01 += base[3 * C + c];
    M11 += base[4 * C + c];
  }
  const float c00 = fmaf(-S0 * invB, S0, M00);
  const float c01 = fmaf(-S0 * invB, S1, M01);
  const float c11 = fmaf(-S1 * invB, S1, M11);
  const float r   = c01 * rsqrtf(c00 * c11);
  fro[c] = 1.41421356237309515f * fabsf(r);   // ||corr - I||_F for 2x2
}

// ---------------------------------------------------------------------------
// Pass 3: single block reduces the 4096 channel losses to the mean.
// Wave-level sum uses V_WMMA_F32_16X16X4_F32 with A = all-ones:
//   D[m,n] = sum_k B[k,n]  =>  sum_{n=0..15} D[0,n] = sum of ALL 64 B slots,
// which is invariant to the (k,n)->slot mapping, so placing each lane's
// partial in B.x (and 0 in B.y) gives a provably correct 32-lane sum.
// D row 0 = d[0] in lanes 0..15 (documented C/D layout).
// ---------------------------------------------------------------------------
__global__ void __launch_bounds__(TPB)
orth_pass3(const float* __restrict__ fro, float* __restrict__ out) {
  const int t = threadIdx.x;
  float acc = 0.f;
#pragma unroll
  for (int i = 0; i < C / TPB; ++i)
    acc += fro[t + i * TPB];

  // Uniform control flow up to here: EXEC is all ones (WMMA requirement).
  v2f a; a.x = 1.f; a.y = 1.f;       // A = ones (layout-independent)
  v2f b; b.x = acc; b.y = 0.f;       // 32 partials live in the 64 B slots
  v8f cz = {};
  v8f d = __builtin_amdgcn_wmma_f32_16x16x4_f32(
      /*neg_a=*/false, a, /*neg_b=*/false, b,
      /*c_mod=*/(short)0, cz, /*reuse_a=*/false, /*reuse_b=*/false);

  // d[0], lanes 0..15 hold the 16 column sums; fold them.
  float colsum = d[0];
  colsum += __shfl_xor(colsum, 8, 16);
  colsum += __shfl_xor(colsum, 4, 16);
  colsum += __shfl_xor(colsum, 2, 16);
  colsum += __shfl_xor(colsum, 1, 16);

  __shared__ float wsum[TPB / 32];
  const int wave = t >> 5, lane = t & 31;
  if (lane == 0) wsum[wave] = colsum;   // lane 0 holds the wave total
  __syncthreads();
  if (t == 0) {
    float tot = 0.f;
#pragma unroll
    for (int w = 0; w < TPB / 32; ++w) tot += wsum[w];
    out[0] = tot * (1.0f / (float)C);
  }
}

// ---------------------------------------------------------------------------
extern "C" void kernel_launch(void* const* d_in, const int* in_sizes, int n_in,
                              void* d_out, int out_size, void* d_ws,
                              size_t ws_size, hipStream_t stream) {
  (void)n_in; (void)out_size;
  const float4* in  = (const float4*)d_in[0];
  float*        out = (float*)d_out;
  float*        ws  = (float*)d_ws;

  const int total = in_sizes[0];          // B * C * 2
  const int B     = total / (C * 2);      // 4096 for this module

  // b-slices: power of two dividing B, sized to fit workspace.
  int NY = 64;
  while (NY > 1 &&
         (((size_t)NY * 5 * C + C) * sizeof(float) > ws_size || (B % NY) != 0))
    NY >>= 1;
  const int rowsPerBlock = B / NY;

  float* partial = ws;                       // [NY][5][C]
  float* fro     = ws + (size_t)NY * 5 * C;  // [C]

  dim3 g1(C / 2 / TPB, NY);                  // (8, NY) blocks, 256 thr each
  orth_pass1<<<g1, TPB, 0, stream>>>(in, partial, rowsPerBlock);
  orth_pass2<<<C / TPB, TPB, 0, stream>>>(partial, fro, NY, 1.0f / (float)B);
  orth_pass3<<<1, TPB, 0, stream>>>(fro, out);
}